// TorchPCEN_15728170237985
// MI455X (gfx1250) — compile-verified
//
#include <hip/hip_runtime.h>

// ---------------------------------------------------------------------------
// PCEN on gfx1250: EMA scan as chained V_WMMA_F32_16X16X4_F32, with
// double-buffered GLOBAL_LOAD_ASYNC_TO_LDS_B128 staging (ASYNCcnt pipeline).
//
//   m[c,t] = (1-s)^(t+1)*carry[c] + sum_{k<=t} s*(1-s)^(t-k) x[c,k]
//   => per 16-t block: M(16x16) = X(16x16)*U(16x16) + C_carry
//   => four K=4 f32 WMMAs, carry folded into the accumulator; carry for the
//      next block = column t=15 of D, broadcast via width-16 shuffle (DPP).
//
// One wave owns 16 channels for the whole T=4000 stream. Time is tiled in
// 80-step chunks; tile i+1's async DMA overlaps tile i's WMMA + PCEN math.
// The tile loop is manually unrolled 2x so each LDS buffer is addressed
// through a distinct constant pointer -> infer-address-space keeps all
// compute-side reads on the DS path (ds_load), not FLAT.
// ---------------------------------------------------------------------------

typedef float v2f __attribute__((ext_vector_type(2)));
typedef float v8f __attribute__((ext_vector_type(8)));

#define T_LEN      4000
#define TILE_T     80            // 5 x 16-step WMMA sub-blocks; divides 4000
#define N_TILES    (T_LEN / TILE_T)      // 50 (even)
#define SUBT       16
#define WAVES      4             // waves per block
#define ROW_PITCH  84            // LDS row pitch (floats): 336B rows, 16B
                                 // aligned; (c*84+t)%64 conflict-free

#define PCEN_S     0.025f
#define PCEN_OMS   0.975f        // 1 - s
#define PCEN_EPS   1e-6f
#define PCEN_ALPHA 0.98f
#define PCEN_DR    1.4142135623730951f   // delta^r = sqrt(2)

// Issue one 16ch x 80t tile as 10 async global->LDS B128 transfers.
// Each instruction moves 32 lanes x 16B = 512B, fully coalesced
// (20 consecutive float4s per channel row). Tracked by ASYNCcnt.
__device__ __forceinline__ void stage_tile_async(unsigned lds_base,
                                                 const float* __restrict__ xrow,
                                                 int t0, int lane)
{
    #pragma unroll
    for (int i = 0; i < 10; ++i) {
        const int q  = i * 32 + lane;        // 0..319
        const int r  = q / 20;               // channel row
        const int c4 = q % 20;               // float4 column
        const unsigned laddr = lds_base + (unsigned)((r * ROW_PITCH + c4 * 4) * 4);
        const float*   gaddr = xrow + (size_t)r * T_LEN + t0 + c4 * 4;
        asm volatile("global_load_async_to_lds_b128 %0, %1, off"
                     :: "v"(laddr), "v"(gaddr));
    }
}

// Process one staged 16ch x 80t tile: 5 sub-blocks x 4 chained WMMAs,
// PCEN nonlinearity, NT store, carry propagation. `slab` must be a
// constant-per-call-site pointer into __shared__ so reads lower to ds_load.
__device__ __forceinline__ void process_tile(int ti,
                                             float* slab,
                                             unsigned lbase_next,
                                             const float* __restrict__ xrow,
                                             float* __restrict__ orow,
                                             const v2f bmat[4],
                                             float pow_t1,
                                             int lane, int tl, int hi,
                                             v8f& carry)
{
    const int t0 = ti * TILE_T;

    // Current tile's async DMA has landed in LDS.
    asm volatile("s_wait_asynccnt 0" ::: "memory");

    if (ti + 1 < N_TILES) {
        // LDS reads of the other buffer (tile ti-1) must be drained before
        // its slab is overwritten by tile ti+1's DMA.
        asm volatile("s_wait_dscnt 0" ::: "memory");
        stage_tile_async(lbase_next, xrow, t0 + TILE_T, lane);
    }

    #pragma unroll
    for (int st = 0; st < 5; ++st) {
        const int tb = st * SUBT;

        v8f acc;                              // C init = carry * (1-s)^(t+1)
        #pragma unroll
        for (int v = 0; v < 8; ++v) acc[v] = carry[v] * pow_t1;

        #pragma unroll
        for (int j = 0; j < 4; ++j) {
            const int k0 = 4 * j + (hi ? 2 : 0);
            // A operand: X[M=tl][K=k0..k0+1] (8B-aligned LDS read)
            const v2f a = *reinterpret_cast<const v2f*>(
                slab + tl * ROW_PITCH + tb + k0);
            acc = __builtin_amdgcn_wmma_f32_16x16x4_f32(
                /*neg_a=*/false, a, /*neg_b=*/false, bmat[j],
                /*c_mod=*/(short)0, acc,
                /*reuse_a=*/false, /*reuse_b=*/false);
        }

        // ---- PCEN nonlinearity + NT store, element (c=v+hi*8, t=tl) -------
        #pragma unroll
        for (int v = 0; v < 8; ++v) {
            const int   ch = v + hi * 8;
            const float xv = slab[ch * ROW_PITCH + tb + tl];
            const float m  = acc[v];
            const float g  = __builtin_amdgcn_exp2f(
                               -PCEN_ALPHA * __builtin_amdgcn_logf(PCEN_EPS + m));
            const float pc = __builtin_amdgcn_sqrtf(xv * g + 2.0f) - PCEN_DR;
            __builtin_nontemporal_store(
                pc, orow + (size_t)ch * T_LEN + t0 + tb + tl);
        }

        // ---- carry = column t=15 of D, broadcast in each half-wave --------
        #pragma unroll
        for (int v = 0; v < 8; ++v)
            carry[v] = __shfl(acc[v], 15, 16);
    }
}

__global__ __launch_bounds__(WAVES * 32)
void pcen_wmma_scan(const float* __restrict__ x, float* __restrict__ out)
{
    __shared__ __align__(16) float lds[WAVES][2][16 * ROW_PITCH];

    const int lane  = threadIdx.x & 31;
    const int wave  = threadIdx.x >> 5;
    const int tl    = lane & 15;       // time column (B/C/D) or channel row (A)
    const int hi    = lane >> 4;       // half-wave (selects K pair)
    const int chan0 = (blockIdx.x * WAVES + wave) << 4;

    // --- per-lane constant tables ------------------------------------------
    float pow_t1 = PCEN_OMS;                       // (1-s)^(tl+1)
    for (int i = 0; i < tl; ++i) pow_t1 *= PCEN_OMS;

    // B operand (constant geometric matrix U), WMMA 4x16 f32 layout:
    //   VGPR0 <-> K = 4j + (hi?2:0), VGPR1 <-> K+1, N = tl
    v2f bmat[4];
    for (int j = 0; j < 4; ++j) {
        const int k0 = 4 * j + (hi ? 2 : 0);
        float b0 = 0.f, b1 = 0.f;
        if (tl >= k0)     { float p = PCEN_S; for (int i = 0; i < tl - k0;     ++i) p *= PCEN_OMS; b0 = p; }
        if (tl >= k0 + 1) { float p = PCEN_S; for (int i = 0; i < tl - k0 - 1; ++i) p *= PCEN_OMS; b1 = p; }
        v2f b; b.x = b0; b.y = b1;
        bmat[j] = b;
    }

    const float* xrow = x   + (size_t)chan0 * T_LEN;
    float*       orow = out + (size_t)chan0 * T_LEN;

    // Distinct constant pointers per buffer (no pointer array / select!)
    float* slab0 = &lds[wave][0][0];
    float* slab1 = &lds[wave][1][0];
    // LDS aperture maps addr[31:0] directly -> low 32 bits of the generic
    // pointer are the LDS byte address for the async-load VDST operand.
    const unsigned lb0 = (unsigned)(uintptr_t)slab0;
    const unsigned lb1 = (unsigned)(uintptr_t)slab1;

    v8f carry;                              // per-channel EMA carry, D-layout
    #pragma unroll
    for (int v = 0; v < 8; ++v) carry[v] = 0.0f;

    // Prologue: kick off tile 0 DMA into buffer 0.
    stage_tile_async(lb0, xrow, 0, lane);

    // 2x-unrolled ping-pong so each buffer keeps a constant pointer.
    for (int ti = 0; ti < N_TILES; ti += 2) {
        process_tile(ti,     slab0, lb1, xrow, orow, bmat, pow_t1,
                     lane, tl, hi, carry);
        process_tile(ti + 1, slab1, lb0, xrow, orow, bmat, pow_t1,
                     lane, tl, hi, carry);
    }
}

// ---------------------------------------------------------------------------
extern "C" void kernel_launch(void* const* d_in, const int* in_sizes, int n_in,
                              void* d_out, int out_size, void* d_ws, size_t ws_size,
                              hipStream_t stream)
{
    (void)n_in; (void)out_size; (void)d_ws; (void)ws_size;
    const float* x    = (const float*)d_in[0];
    float*       outp = (float*)d_out;

    const int nchan  = in_sizes[0] / T_LEN;          // 64*128 = 8192
    const int blocks = nchan / (16 * WAVES);         // 128
    pcen_wmma_scan<<<blocks, WAVES * 32, 0, stream>>>(x, outp);
}